// QuasiNewton_7232724927246
// MI455X (gfx1250) — compile-verified
//
#include <hip/hip_runtime.h>

#define NDIM 32
#define WAVES_PER_BLOCK 8
#define WAVE_SMEM (1024 + 64)
#define RO 0.5f
#define CARM 1e-4f
#define MAX_BT 25

typedef __attribute__((ext_vector_type(2))) float v2f;
typedef __attribute__((ext_vector_type(8))) float v8f;

__device__ __forceinline__ float wave_sum(float x) {
#pragma unroll
  for (int off = 16; off > 0; off >>= 1) x += __shfl_xor(x, off, 32);
  return x;
}

// Async DMA of 16 bytes global -> LDS (per lane), tracked by ASYNCcnt.
__device__ __forceinline__ void async_copy16(unsigned lds_off, const void* gptr) {
  asm volatile("global_load_async_to_lds_b128 %0, %1, off" ::"v"(lds_off),
               "v"((unsigned long long)(size_t)gptr)
               : "memory");
}
__device__ __forceinline__ void wait_async0() {
  asm volatile("s_wait_asynccnt 0" ::: "memory");
}
__device__ __forceinline__ void wait_ds0() {
  asm volatile("s_wait_dscnt 0" ::: "memory");
}

// u = M @ v, M 32x32 as 2x2 grid of 16x16 WMMA-A blocks (4 K-chunks each).
// v one element per lane; result one element per lane. vecbuf: >=32 floats LDS.
__device__ __forceinline__ float wmma_matvec(const v2f A[2][2][4], float v,
                                             float* vecbuf, int lane) {
  vecbuf[lane] = v;
  wait_ds0();
  const int sel2 = (lane >> 4) * 2;  // lanes>=16 carry K{2,3}
  const bool carrier = ((lane & 15) == 0);
  v2f Bq[8];
#pragma unroll
  for (int q = 0; q < 8; ++q) {
    v2f t = *reinterpret_cast<const v2f*>(&vecbuf[q * 4 + sel2]);
    Bq[q].x = carrier ? t.x : 0.0f;  // only column N=0 nonzero
    Bq[q].y = carrier ? t.y : 0.0f;
  }
  v8f acc0 = {};
  v8f acc1 = {};
#pragma unroll
  for (int bj = 0; bj < 2; ++bj) {
#pragma unroll
    for (int c = 0; c < 4; ++c) {
      const v2f B = Bq[bj * 4 + c];
      acc0 = __builtin_amdgcn_wmma_f32_16x16x4_f32(false, A[0][bj][c], false, B,
                                                   (short)0, acc0, false, false);
      acc1 = __builtin_amdgcn_wmma_f32_16x16x4_f32(false, A[1][bj][c], false, B,
                                                   (short)0, acc1, false, false);
    }
  }
  // D column N=0: lane 0 holds rows 0..7 (vgpr idx), lane 16 rows 8..15.
  if (carrier) {
    const int off = (lane >> 4) * 8;
    float4 a0, a1, b0, b1;
    a0.x = acc0[0]; a0.y = acc0[1]; a0.z = acc0[2]; a0.w = acc0[3];
    a1.x = acc0[4]; a1.y = acc0[5]; a1.z = acc0[6]; a1.w = acc0[7];
    b0.x = acc1[0]; b0.y = acc1[1]; b0.z = acc1[2]; b0.w = acc1[3];
    b1.x = acc1[4]; b1.y = acc1[5]; b1.z = acc1[6]; b1.w = acc1[7];
    *reinterpret_cast<float4*>(&vecbuf[off]) = a0;
    *reinterpret_cast<float4*>(&vecbuf[off + 4]) = a1;
    *reinterpret_cast<float4*>(&vecbuf[16 + off]) = b0;
    *reinterpret_cast<float4*>(&vecbuf[16 + off + 4]) = b1;
  }
  wait_ds0();
  return vecbuf[lane];
}

// u = h_k @ v with h_k row `lane` in registers; broadcasts via b128 LDS loads.
__device__ __forceinline__ float hk_matvec(const float hk[32], float v,
                                           float* vecbuf, int lane) {
  vecbuf[lane] = v;
  wait_ds0();
  float acc = 0.0f;
#pragma unroll
  for (int t = 0; t < 8; ++t) {
    float4 c4 = *reinterpret_cast<const float4*>(&vecbuf[t * 4]);
    acc = fmaf(hk[t * 4 + 0], c4.x, acc);
    acc = fmaf(hk[t * 4 + 1], c4.y, acc);
    acc = fmaf(hk[t * 4 + 2], c4.z, acc);
    acc = fmaf(hk[t * 4 + 3], c4.w, acc);
  }
  return acc;
}

__global__ __launch_bounds__(256) void bfgs_wmma_kernel(
    const float* __restrict__ yv, const float* __restrict__ h,
    const float* __restrict__ x0, const float* __restrict__ alpha0,
    const float* __restrict__ hk0, const int* __restrict__ iters_p,
    float* __restrict__ out, int nprob) {
  __shared__ float smem[WAVES_PER_BLOCK * WAVE_SMEM];
  const int lane = threadIdx.x & 31;
  const int wid = threadIdx.x >> 5;
  const int b = blockIdx.x * WAVES_PER_BLOCK + wid;
  if (b >= nprob) return;
  float* mat = &smem[wid * WAVE_SMEM];
  float* vecbuf = mat + 1024;
  // Flat LDS address low 32 bits == LDS byte offset (aperture rule).
  const unsigned mat_lds = (unsigned)(size_t)mat;

  const int iters = iters_p[0];

  // ---- stage h (4 KB) into LDS via async DMA ----
  {
    const float4* g4 = reinterpret_cast<const float4*>(h + (size_t)b * 1024);
#pragma unroll
    for (int t = 0; t < 8; ++t)
      async_copy16(mat_lds + (unsigned)(t * 32 + lane) * 16, g4 + t * 32 + lane);
  }
  wait_async0();

  // ---- build WMMA A-layout registers for h and h^T (resident all iters) ----
  v2f HA[2][2][4];
  v2f HTA[2][2][4];
  const int mrow = lane & 15;
  const int khalf = (lane >> 4) * 2;  // lanes 0-15: K{0,1}; lanes 16-31: K{2,3}
#pragma unroll
  for (int bi = 0; bi < 2; ++bi)
#pragma unroll
    for (int bj = 0; bj < 2; ++bj)
#pragma unroll
      for (int c = 0; c < 4; ++c) {
        int r = bi * 16 + mrow;
        int c0 = bj * 16 + c * 4 + khalf;
        v2f a;
        a.x = mat[r * 32 + c0];
        a.y = mat[r * 32 + c0 + 1];
        HA[bi][bj][c] = a;
        // h^T block (bi,bj): element (M=mrow, K) = h[bj*16+K][bi*16+mrow]
        int rr = bj * 16 + c * 4 + khalf;
        int cc = bi * 16 + mrow;
        v2f at;
        at.x = mat[rr * 32 + cc];
        at.y = mat[(rr + 1) * 32 + cc];
        HTA[bi][bj][c] = at;
      }
  wait_ds0();  // h reads complete before h_k overwrites the buffer

  // ---- stage h_k via async DMA, keep row `lane` in registers ----
  {
    const float4* g4 = reinterpret_cast<const float4*>(hk0 + (size_t)b * 1024);
#pragma unroll
    for (int t = 0; t < 8; ++t)
      async_copy16(mat_lds + (unsigned)(t * 32 + lane) * 16, g4 + t * 32 + lane);
  }
  wait_async0();
  float hk[32];
#pragma unroll
  for (int c = 0; c < 32; ++c) hk[c] = mat[lane * 32 + c];

  // ---- per-lane vector state ----
  float yl = yv[(size_t)b * 32 + lane];
  float xl = x0[(size_t)b * 32 + lane];
  float alpha = alpha0[b];

  float hx = wmma_matvec(HA, xl, vecbuf, lane);  // h @ x (maintained affinely)
  float r = hx - yl;                             // residual h@x - y
  float g = wmma_matvec(HTA, r, vecbuf, lane);   // gradient h^T r

  for (int it = 0; it < iters; ++it) {
    float p = -hk_matvec(hk, g, vecbuf, lane);   // search direction
    float hp = wmma_matvec(HA, p, vecbuf, lane);

    // scalars for the affine Armijo test: ||r + a*hp||^2 = rr + 2a*rhp + a^2*hphp
    float rr = wave_sum(r * r);
    float rhp = wave_sum(r * hp);
    float hphp = wave_sum(hp * hp);
    float gp = wave_sum(g * p);
    float fx = sqrtf(rr) * 0.5f;

    for (int bt = 0; bt < MAX_BT; ++bt) {
      float ss = fmaf(alpha, fmaf(alpha, hphp, 2.0f * rhp), rr);
      float lhs = sqrtf(ss) * 0.5f;
      float rhs = fmaf(CARM * alpha, gp, fx);
      alpha = (lhs > rhs) ? (RO * alpha) : alpha;
    }

    float s = alpha * p;
    xl += s;
    hx = fmaf(alpha, hp, hx);
    r = hx - yl;
    float gn = wmma_matvec(HTA, r, vecbuf, lane);  // new gradient
    float yk = gn - g;

    float hy = hk_matvec(hk, yk, vecbuf, lane);    // h_k @ y_k (symmetric)

    float auxSc = wave_sum(s * yk);
    float ytHy = wave_sum(yk * hy);
    float inv = 1.0f / auxSc;
    float a = (auxSc + ytHy) * inv * inv;

    // h_k += a*(s s^T) - (hy s^T + s hy^T)/auxSc   (row `lane`)
    vecbuf[lane] = s;
    vecbuf[32 + lane] = hy;
    wait_ds0();
#pragma unroll
    for (int t = 0; t < 8; ++t) {
      float4 sc = *reinterpret_cast<const float4*>(&vecbuf[t * 4]);
      float4 hc = *reinterpret_cast<const float4*>(&vecbuf[32 + t * 4]);
      hk[t * 4 + 0] += a * s * sc.x - inv * (hy * sc.x + s * hc.x);
      hk[t * 4 + 1] += a * s * sc.y - inv * (hy * sc.y + s * hc.y);
      hk[t * 4 + 2] += a * s * sc.z - inv * (hy * sc.z + s * hc.z);
      hk[t * 4 + 3] += a * s * sc.w - inv * (hy * sc.w + s * hc.w);
    }
    g = gn;
  }

  out[(size_t)b * 32 + lane] = xl;
}

extern "C" void kernel_launch(void* const* d_in, const int* in_sizes, int n_in,
                              void* d_out, int out_size, void* d_ws, size_t ws_size,
                              hipStream_t stream) {
  const float* yv = (const float*)d_in[0];
  const float* h = (const float*)d_in[1];
  const float* x = (const float*)d_in[2];
  const float* alpha = (const float*)d_in[3];
  const float* hk = (const float*)d_in[4];
  const int* iters = (const int*)d_in[5];
  float* out = (float*)d_out;
  int nprob = in_sizes[0] / NDIM;  // B
  int blocks = (nprob + WAVES_PER_BLOCK - 1) / WAVES_PER_BLOCK;
  bfgs_wmma_kernel<<<blocks, 32 * WAVES_PER_BLOCK, 0, stream>>>(
      yv, h, x, alpha, hk, iters, out, nprob);
}